// Extractor_51591147160040
// MI455X (gfx1250) — compile-verified
//
#include <hip/hip_runtime.h>
#include <hip/hip_bf16.h>

#define NMS_RADIUS 4
#define THRESHOLD 0.005f
#define BORDER 4
#define MAX_KP 1024
#define CAND_CAP 8192
#define NKP 1024   // keypoint count == GEMM N dimension, compile-time constant

typedef __attribute__((ext_vector_type(2))) float v2f;
typedef __attribute__((ext_vector_type(8))) float v8f;

// ---------------------------------------------------------------------------
// NMS: separable 9x9 max-pool passes + elementwise mask logic
// ---------------------------------------------------------------------------
__global__ void mp_h_kernel(const float* __restrict__ in, float* __restrict__ out,
                            int H, int W) {
    int x = blockIdx.x * blockDim.x + threadIdx.x;
    int y = blockIdx.y;
    int b = blockIdx.z;
    if (x >= W) return;
    const float* row = in + ((size_t)b * H + y) * W;
    float m = -3.402823466e+38f;
    #pragma unroll
    for (int dx = -NMS_RADIUS; dx <= NMS_RADIUS; ++dx) {
        int xx = x + dx;
        if (xx >= 0 && xx < W) m = fmaxf(m, row[xx]);
    }
    out[((size_t)b * H + y) * W + x] = m;
}

__global__ void mp_v_kernel(const float* __restrict__ in, float* __restrict__ out,
                            int H, int W) {
    int x = blockIdx.x * blockDim.x + threadIdx.x;
    int y = blockIdx.y;
    int b = blockIdx.z;
    if (x >= W) return;
    const float* img = in + (size_t)b * H * W;
    float m = -3.402823466e+38f;
    #pragma unroll
    for (int dy = -NMS_RADIUS; dy <= NMS_RADIUS; ++dy) {
        int yy = y + dy;
        if (yy >= 0 && yy < H) m = fmaxf(m, img[(size_t)yy * W + x]);
    }
    out[((size_t)b * H + y) * W + x] = m;
}

__global__ void eq_mask_kernel(const float* __restrict__ S, const float* __restrict__ pool,
                               float* __restrict__ mask, size_t total) {
    size_t i = (size_t)blockIdx.x * blockDim.x + threadIdx.x;
    if (i >= total) return;
    mask[i] = (S[i] == pool[i]) ? 1.0f : 0.0f;
}

__global__ void supp_kernel(const float* __restrict__ S, const float* __restrict__ poolm,
                            float* __restrict__ supp, size_t total) {
    size_t i = (size_t)blockIdx.x * blockDim.x + threadIdx.x;
    if (i >= total) return;
    supp[i] = (poolm[i] > 0.0f) ? 0.0f : S[i];
}

__global__ void upd_mask_kernel(float* __restrict__ mask, const float* __restrict__ supp,
                                const float* __restrict__ pools, const float* __restrict__ poolm,
                                size_t total) {
    size_t i = (size_t)blockIdx.x * blockDim.x + threadIdx.x;
    if (i >= total) return;
    bool supp_mask = poolm[i] > 0.0f;
    bool new_max   = (supp[i] == pools[i]);
    bool m = (mask[i] != 0.0f) || (new_max && !supp_mask);
    mask[i] = m ? 1.0f : 0.0f;
}

// ---------------------------------------------------------------------------
// Candidate compaction (mask & threshold & border)
// ---------------------------------------------------------------------------
__global__ void compact_kernel(const float* __restrict__ S, const float* __restrict__ mask,
                               int* __restrict__ cnt, float* __restrict__ cs,
                               int* __restrict__ ci, int H, int W) {
    int x = blockIdx.x * blockDim.x + threadIdx.x;
    int y = blockIdx.y;
    int b = blockIdx.z;
    if (x >= W) return;
    size_t i = ((size_t)b * H + y) * W + x;
    float sv = (mask[i] != 0.0f) ? S[i] : 0.0f;
    bool valid = (y >= BORDER) && (y < H - BORDER) && (x >= BORDER) && (x < W - BORDER);
    if (valid && sv > THRESHOLD) {
        int p = atomicAdd(&cnt[b], 1);
        if (p < CAND_CAP) {
            cs[(size_t)b * CAND_CAP + p] = sv;
            ci[(size_t)b * CAND_CAP + p] = y * W + x;
        }
    }
}

// ---------------------------------------------------------------------------
// Per-batch in-LDS bitonic sort (descending), emit kp + top_scores
// ---------------------------------------------------------------------------
__global__ void __launch_bounds__(512)
topk_sort_kernel(const int* __restrict__ cnt, const float* __restrict__ cs,
                 const int* __restrict__ ci, float* __restrict__ kp,
                 float* __restrict__ tsc, int W, int N) {
    __shared__ float sc[CAND_CAP];
    __shared__ int   id[CAND_CAP];
    int b = blockIdx.x;
    int t = threadIdx.x;
    const int T = blockDim.x;
    int count = cnt[b];
    if (count > CAND_CAP) count = CAND_CAP;
    for (int i = t; i < CAND_CAP; i += T) {
        if (i < count) {
            sc[i] = cs[(size_t)b * CAND_CAP + i];
            id[i] = ci[(size_t)b * CAND_CAP + i];
        } else {
            sc[i] = -1.0f;  // matches reference masked fill value
            id[i] = i;
        }
    }
    __syncthreads();
    for (int k = 2; k <= CAND_CAP; k <<= 1) {
        for (int j = k >> 1; j > 0; j >>= 1) {
            for (int i = t; i < CAND_CAP; i += T) {
                int ixj = i ^ j;
                if (ixj > i) {
                    bool desc_seg = ((i & k) == 0);
                    float si = sc[i], sj = sc[ixj];
                    int ii = id[i], ij = id[ixj];
                    bool jFirst = (sj > si) || (sj == si && ij < ii);
                    if (jFirst == desc_seg) {
                        sc[i] = sj; sc[ixj] = si;
                        id[i] = ij; id[ixj] = ii;
                    }
                }
            }
            __syncthreads();
        }
    }
    for (int n = t; n < N; n += T) {
        float s = sc[n];
        int idx = id[n];
        tsc[(size_t)b * N + n] = s;
        kp[((size_t)b * N + n) * 2 + 0] = (float)(idx % W);
        kp[((size_t)b * N + n) * 2 + 1] = (float)(idx / W);
    }
}

// ---------------------------------------------------------------------------
// Bilinear descriptor sampling + L2 normalize. One block per (kp, batch),
// one thread per channel. Output layout [b][c][n] (channel-major for GEMM-B).
// ---------------------------------------------------------------------------
__global__ void sample_desc_kernel(const float* __restrict__ desc, int C, int h, int w,
                                   float s, const float* __restrict__ kp,
                                   float* __restrict__ out, long outBatchStride, int N) {
    __shared__ float red[256];
    __shared__ float s_inv;
    int n = blockIdx.x;
    int b = blockIdx.y;
    int c = threadIdx.x;

    float kx = kp[((size_t)b * N + n) * 2 + 0];
    float ky = kp[((size_t)b * N + n) * 2 + 1];
    // k = kp - s/2 + 0.5 ; x = k/(w*s-1)*(w-1)  (align_corners=True grid_sample)
    float fx = (kx - s * 0.5f + 0.5f) / (w * s - 1.0f) * (float)(w - 1);
    float fy = (ky - s * 0.5f + 0.5f) / (h * s - 1.0f) * (float)(h - 1);
    float x0f = floorf(fx), y0f = floorf(fy);
    float wx1 = fx - x0f, wy1 = fy - y0f;
    float wx0 = 1.0f - wx1, wy0 = 1.0f - wy1;
    int x0 = (int)x0f, y0 = (int)y0f;

    const float* db = desc + ((size_t)b * C + c) * h * w;

    auto tap = [&](int xi, int yi) -> float {
        float v = (xi >= 0 && xi < w && yi >= 0 && yi < h) ? 1.0f : 0.0f;
        int xc = xi < 0 ? 0 : (xi > w - 1 ? w - 1 : xi);
        int yc = yi < 0 ? 0 : (yi > h - 1 ? h - 1 : yi);
        return db[(size_t)yc * w + xc] * v;
    };

    float acc = tap(x0, y0) * (wx0 * wy0) + tap(x0 + 1, y0) * (wx1 * wy0)
              + tap(x0, y0 + 1) * (wx0 * wy1) + tap(x0 + 1, y0 + 1) * (wx1 * wy1);

    red[c] = acc * acc;
    __syncthreads();
    for (int st = blockDim.x >> 1; st > 0; st >>= 1) {
        if (c < st) red[c] += red[c + st];
        __syncthreads();
    }
    if (c == 0) s_inv = 1.0f / fmaxf(sqrtf(red[0]), 1e-12f);
    __syncthreads();

    out[(size_t)b * outBatchStride + (size_t)c * N + n] = acc * s_inv;
}

// ---------------------------------------------------------------------------
// f32 GEMM on the matrix pipe: Y[o,n] = sum_k W[o,k]*X[k,n] + bias[o]
// One wave computes a 16x64 output tile: 4 accumulators share each A
// fragment, 4x V_WMMA_F32_16X16X4_F32 per K-step. N is compile-time (1024)
// so all B-operand loads are immediate-offset off a single pointer that
// advances by a constant 4*N floats per step (no per-iteration multiplies).
// A (16x4 f32): lanes 0-15 hold K={k0,k0+1}, lanes 16-31 hold K={k0+2,k0+3}.
// ---------------------------------------------------------------------------
__global__ void __launch_bounds__(32)
gemm_wmma_f32_kernel(const float* __restrict__ Wm, const float* __restrict__ bias,
                     const float* __restrict__ X, float* __restrict__ Y,
                     int K, long xBatchStride, long yBatchStride) {
    const int N = NKP;
    int lane = threadIdx.x;
    int l = lane & 15;
    int half = lane >> 4;
    int m0 = blockIdx.x * 16;
    int n0 = blockIdx.y * 64;
    const float* Xb = X + (size_t)blockIdx.z * xBatchStride;
    float* Yb = Y + (size_t)blockIdx.z * yBatchStride;

    // per-lane streaming pointers
    const float* aPtr = Wm + (size_t)(m0 + l) * K + 2 * half;     // [0],[1] -> b64
    const float* bPtr = Xb + (size_t)(2 * half) * N + n0 + l;     // rows ka, ka+1

    v8f acc0 = {}, acc1 = {}, acc2 = {}, acc3 = {};
    for (int k0 = 0; k0 < K; k0 += 4) {
        v2f a;
        a.x = aPtr[0];
        a.y = aPtr[1];
        v2f b0, b1, b2, b3;
        b0.x = bPtr[0];      b0.y = bPtr[N];
        b1.x = bPtr[16];     b1.y = bPtr[N + 16];
        b2.x = bPtr[32];     b2.y = bPtr[N + 32];
        b3.x = bPtr[48];     b3.y = bPtr[N + 48];
        // speculative streaming prefetch of the next K-block's B rows
        // (non-faulting per ISA: translation failures silently dropped)
        __builtin_prefetch(&bPtr[4 * N], 0, 0);
        acc0 = __builtin_amdgcn_wmma_f32_16x16x4_f32(false, a, false, b0, (short)0, acc0, false, false);
        acc1 = __builtin_amdgcn_wmma_f32_16x16x4_f32(false, a, false, b1, (short)0, acc1, false, false);
        acc2 = __builtin_amdgcn_wmma_f32_16x16x4_f32(false, a, false, b2, (short)0, acc2, false, false);
        acc3 = __builtin_amdgcn_wmma_f32_16x16x4_f32(false, a, false, b3, (short)0, acc3, false, false);
        aPtr += 4;
        bPtr += 4 * N;
    }

    #pragma unroll
    for (int r = 0; r < 8; ++r) {
        int m = m0 + r + 8 * half;
        float bv = bias[m];
        float* yRow = Yb + (size_t)m * N + n0 + l;
        yRow[0]  = acc0[r] + bv;
        yRow[16] = acc1[r] + bv;
        yRow[32] = acc2[r] + bv;
        yRow[48] = acc3[r] + bv;
    }
}

// ---------------------------------------------------------------------------
extern "C" void kernel_launch(void* const* d_in, const int* in_sizes, int n_in,
                              void* d_out, int out_size, void* d_ws, size_t ws_size,
                              hipStream_t stream) {
    const int B = 4, H = 480, W = 640, N = MAX_KP;
    const float* scores  = (const float*)d_in[0];
    const float* d1      = (const float*)d_in[1];   // [B,64,240,320]
    const float* d2      = (const float*)d_in[2];   // [B,128,120,160]
    const float* cDa     = (const float*)d_in[3];   // [B,256,60,80]
    const float* d4      = (const float*)d_in[4];   // [B,256,30,40]
    const float* lin0_w  = (const float*)d_in[5];   // [256,256]
    const float* lin0_b  = (const float*)d_in[6];
    const float* lin1_w  = (const float*)d_in[7];   // [128,128]
    const float* lin1_b  = (const float*)d_in[8];
    const float* lin2_w  = (const float*)d_in[9];   // [64,64]
    const float* lin2_b  = (const float*)d_in[10];
    const float* merge_w = (const float*)d_in[11];  // [256,704]
    const float* merge_b = (const float*)d_in[12];

    float* kp_out  = (float*)d_out;                    // [B,N,2]
    float* tsc_out = kp_out + (size_t)B * N * 2;       // [B,N]
    float* desc_out = tsc_out + (size_t)B * N;         // [B,256,N]

    // workspace carve-out
    size_t HW = (size_t)B * H * W;
    float* hbuf  = (float*)d_ws;
    float* pool  = hbuf  + HW;
    float* mask  = pool  + HW;
    float* poolm = mask  + HW;
    float* supp  = poolm + HW;
    float* pools = supp  + HW;
    int*   cnt   = (int*)(pools + HW);
    float* cs    = (float*)(cnt + 16);
    int*   ci    = (int*)(cs + (size_t)B * CAND_CAP);
    float* de1s  = (float*)(ci + (size_t)B * CAND_CAP);          // [B,64,N]
    float* de2s  = de1s + (size_t)B * 64 * N;                    // [B,128,N]
    float* de4s  = de2s + (size_t)B * 128 * N;                   // [B,256,N]
    float* cat   = de4s + (size_t)B * 256 * N;                   // [B,704,N]

    dim3 pix_grid((W + 255) / 256, H, B);
    dim3 pix_blk(256);
    size_t total = HW;
    int eb = (int)((total + 255) / 256);

    // simple_nms
    mp_h_kernel<<<pix_grid, pix_blk, 0, stream>>>(scores, hbuf, H, W);
    mp_v_kernel<<<pix_grid, pix_blk, 0, stream>>>(hbuf, pool, H, W);
    eq_mask_kernel<<<eb, 256, 0, stream>>>(scores, pool, mask, total);
    for (int it = 0; it < 2; ++it) {
        mp_h_kernel<<<pix_grid, pix_blk, 0, stream>>>(mask, hbuf, H, W);
        mp_v_kernel<<<pix_grid, pix_blk, 0, stream>>>(hbuf, poolm, H, W);
        supp_kernel<<<eb, 256, 0, stream>>>(scores, poolm, supp, total);
        mp_h_kernel<<<pix_grid, pix_blk, 0, stream>>>(supp, hbuf, H, W);
        mp_v_kernel<<<pix_grid, pix_blk, 0, stream>>>(hbuf, pools, H, W);
        upd_mask_kernel<<<eb, 256, 0, stream>>>(mask, supp, pools, poolm, total);
    }

    // top-k via compact + per-batch bitonic sort
    hipMemsetAsync(cnt, 0, 16 * sizeof(int), stream);
    compact_kernel<<<pix_grid, pix_blk, 0, stream>>>(scores, mask, cnt, cs, ci, H, W);
    topk_sort_kernel<<<B, 512, 0, stream>>>(cnt, cs, ci, kp_out, tsc_out, W, N);

    // descriptor sampling (de3 lands directly in concat rows 192..447)
    dim3 sg(N, B);
    sample_desc_kernel<<<sg, 64,  0, stream>>>(d1, 64, 240, 320, 2.0f, kp_out,
                                               de1s, (long)64 * N, N);
    sample_desc_kernel<<<sg, 128, 0, stream>>>(d2, 128, 120, 160, 4.0f, kp_out,
                                               de2s, (long)128 * N, N);
    sample_desc_kernel<<<sg, 256, 0, stream>>>(cDa, 256, 60, 80, 8.0f, kp_out,
                                               cat + (size_t)192 * N, (long)704 * N, N);
    sample_desc_kernel<<<sg, 256, 0, stream>>>(d4, 256, 30, 40, 16.0f, kp_out,
                                               de4s, (long)256 * N, N);

    // per-scale 1x1 convs (WMMA f32) into concat rows, then merge -> d_out
    gemm_wmma_f32_kernel<<<dim3(64 / 16, NKP / 64, B), 32, 0, stream>>>(
        lin2_w, lin2_b, de1s, cat + (size_t)0 * N, 64, (long)64 * N, (long)704 * N);
    gemm_wmma_f32_kernel<<<dim3(128 / 16, NKP / 64, B), 32, 0, stream>>>(
        lin1_w, lin1_b, de2s, cat + (size_t)64 * N, 128, (long)128 * N, (long)704 * N);
    gemm_wmma_f32_kernel<<<dim3(256 / 16, NKP / 64, B), 32, 0, stream>>>(
        lin0_w, lin0_b, de4s, cat + (size_t)448 * N, 256, (long)256 * N, (long)704 * N);
    gemm_wmma_f32_kernel<<<dim3(256 / 16, NKP / 64, B), 32, 0, stream>>>(
        merge_w, merge_b, cat, desc_out, 704, (long)704 * N, (long)256 * N);
}